// VMFLoss_38036230373447
// MI455X (gfx1250) — compile-verified
//
#include <hip/hip_runtime.h>
#include <hip/hip_bf16.h>

#define KC 128          // K_CLASSES
#define DD 512          // D
#define NR 65536        // N

typedef __attribute__((ext_vector_type(2))) float v2f;
typedef __attribute__((ext_vector_type(8))) float v8f;

// Workspace layout (float offsets):
//   [0      , 65536)  r   (K x D resultant sums)
//   [65536  , 65664)  n   (K counts)
//   [65664  , 65792)  kappa
//   [65792  , 65920)  log_k
//   [65920  , 66048)  bes
//   [66048  , 131584) mu  (K x D)
#define WS_R    0
#define WS_N    65536
#define WS_KAP  65664
#define WS_LGK  65792
#define WS_BES  65920
#define WS_MU   66048

// ---------------------------------------------------------------- zero
__global__ __launch_bounds__(256) void vmf_zero(float* ws, float* out) {
    int i = blockIdx.x * 256 + threadIdx.x;
    if (i < WS_N + KC) ws[i] = 0.0f;      // zero r and n
    if (i == 0) out[0] = 0.0f;
}

// ---------------------------------------------------------------- segment sum
// grid = (8 column-chunks of 64, 32 row-chunks of 2048), block = 256
__global__ __launch_bounds__(256) void vmf_segsum(const float* __restrict__ X,
                                                  const int* __restrict__ y,
                                                  float* __restrict__ r,
                                                  float* __restrict__ n) {
    __shared__ float acc[KC * 64];       // 32 KB per-class partial sums
    __shared__ float hist[KC];
    const int cc  = blockIdx.x;          // column chunk 0..7
    const int rc  = blockIdx.y;          // row chunk 0..31
    const int tid = threadIdx.x;

    for (int i = tid; i < KC * 64; i += 256) acc[i] = 0.0f;
    if (cc == 0) { if (tid < KC) hist[tid] = 0.0f; }
    __syncthreads();

    const int colBase = cc * 64;
    const int c4      = (tid & 15) * 4;  // 4 columns per thread
    const int rOff    = tid >> 4;        // 16 rows per iteration
    const int rowBase = rc * 2048;

    for (int it = 0; it < 128; ++it) {
        const int row = rowBase + it * 16 + rOff;
        const int cls = y[row];
        const float4 xv =
            *reinterpret_cast<const float4*>(&X[(size_t)row * DD + colBase + c4]);
        if (it < 127)  // speculative prefetch of next 16-row slab (global_prefetch_b8)
            __builtin_prefetch(&X[(size_t)(row + 16) * DD + colBase + c4], 0, 0);
        float* a = &acc[cls * 64 + c4];
        atomicAdd(a + 0, xv.x);
        atomicAdd(a + 1, xv.y);
        atomicAdd(a + 2, xv.z);
        atomicAdd(a + 3, xv.w);
        if (cc == 0 && (tid & 15) == 0) atomicAdd(&hist[cls], 1.0f);
    }
    __syncthreads();

    for (int i = tid; i < KC * 64; i += 256)
        atomicAdd(&r[(size_t)(i >> 6) * DD + colBase + (i & 63)], acc[i]);
    if (cc == 0 && tid < KC) atomicAdd(&n[tid], hist[tid]);
}

// ---------------------------------------------------------------- per-class stats
// one block, 128 threads (thread == class)
__global__ __launch_bounds__(128) void vmf_stats(const float* __restrict__ r,
                                                 const float* __restrict__ n,
                                                 float* __restrict__ kap,
                                                 float* __restrict__ lgk,
                                                 float* __restrict__ bes,
                                                 float* __restrict__ mu) {
    const int k = threadIdx.x;
    const float4* rp = reinterpret_cast<const float4*>(r + (size_t)k * DD);
    float4*       mp = reinterpret_cast<float4*>(mu + (size_t)k * DD);

    float ss = 0.0f;
    for (int i = 0; i < DD / 4; ++i) {
        float4 v = rp[i];
        ss += v.x * v.x + v.y * v.y + v.z * v.z + v.w * v.w;
    }
    const float norm = sqrtf(ss);
    const float inv  = 1.0f / norm;
    for (int i = 0; i < DD / 4; ++i) {
        float4 v = rp[i];
        v.x *= inv; v.y *= inv; v.z *= inv; v.w *= inv;
        mp[i] = v;
    }

    const float rb = norm / n[k];
    float kappa = ((float)DD * rb - rb * rb * rb) / (1.0f - rb * rb);
    if (rb > 0.9f) kappa = -0.4f + 1.39f * rb + 0.43f / (1.0f - rb);
    kap[k] = kappa;
    lgk[k] = logf(kappa);

    // _ive(d_star, kappa): uniform asymptotic expansion, d_star = 513/2 - 1
    const float v  = 255.5f;
    const float zz = kappa / v;
    const float s  = sqrtf(1.0f + zz * zz);
    const float t  = 1.0f / s;
    const float t2 = t * t;
    const float eta = s + logf(zz / (1.0f + s));
    const float u1 = (3.0f * t - 5.0f * t * t2) / 24.0f;
    const float u2 = (81.0f * t2 - 462.0f * t2 * t2 + 385.0f * t2 * t2 * t2) / 1152.0f;
    const float u3 = (30375.0f * t * t2 - 369603.0f * t * t2 * t2 +
                      765765.0f * t * t2 * t2 * t2 -
                      425425.0f * t * t2 * t2 * t2 * t2) / 414720.0f;
    const float series = 1.0f + u1 / v + u2 / (v * v) + u3 / (v * v * v);
    bes[k] = expf(v * eta - kappa) * series /
             (sqrtf(2.0f * 3.14159265358979f * v) * sqrtf(s));
}

// ---------------------------------------------------------------- WMMA dots + fused kl^2 sum
// grid = (8, 8) 16x16 output tiles, block = 32 (one wave, EXEC all ones)
__global__ __launch_bounds__(32) void vmf_kl(const float* __restrict__ mu,
                                             const float* __restrict__ kap,
                                             const float* __restrict__ lgk,
                                             const float* __restrict__ bes,
                                             float* __restrict__ out) {
    const int tj   = blockIdx.x;
    const int ti   = blockIdx.y;
    const int lane = threadIdx.x;
    const int half = lane >> 4;          // 0: K={d,d+1}, 1: K={d+2,d+3}
    const int l15  = lane & 15;

    // A lane holds row (ti*16+l15) of mu; B lane holds row (tj*16+l15) of mu.
    // dots = G * G^T, so A- and B-fragments use the identical gather pattern;
    // any K permutation consistent between A and B yields correct dot products.
    const float* aRow = mu + (size_t)(ti * 16 + l15) * DD + half * 2;
    const float* bRow = mu + (size_t)(tj * 16 + l15) * DD + half * 2;

    v8f c = {0.f, 0.f, 0.f, 0.f, 0.f, 0.f, 0.f, 0.f};
    for (int d = 0; d < DD; d += 4) {
        v2f a = *reinterpret_cast<const v2f*>(aRow + d);
        v2f b = *reinterpret_cast<const v2f*>(bRow + d);
        c = __builtin_amdgcn_wmma_f32_16x16x4_f32(false, a, false, b,
                                                  (short)0, c, false, false);
    }

    // C/D layout: VGPR v, lanes 0-15 -> M=v, lanes 16-31 -> M=v+8, N=lane&15
    const float dstar = 255.5f;
    const int   j  = tj * 16 + l15;
    const float kj = kap[j], lj = lgk[j], bj = bes[j];
    float accv = 0.0f;
#pragma unroll
    for (int v = 0; v < 8; ++v) {
        const int   i  = ti * 16 + v + half * 8;
        const float kl = dstar * (lj - lgk[i]) - kap[i] + bes[i] - bj + kj * c[v];
        accv += kl * kl;
    }
    for (int off = 16; off > 0; off >>= 1)
        accv += __shfl_down(accv, off, 32);
    if (lane == 0)
        atomicAdd(out, accv * (1.0f / ((float)KC * (float)KC)));
}

// ---------------------------------------------------------------- launch
extern "C" void kernel_launch(void* const* d_in, const int* in_sizes, int n_in,
                              void* d_out, int out_size, void* d_ws, size_t ws_size,
                              hipStream_t stream) {
    (void)in_sizes; (void)n_in; (void)out_size; (void)ws_size;
    const float* X  = (const float*)d_in[0];
    const int*   y  = (const int*)d_in[1];
    float*       ws = (float*)d_ws;
    float*       out = (float*)d_out;

    float* r   = ws + WS_R;
    float* n   = ws + WS_N;
    float* kap = ws + WS_KAP;
    float* lgk = ws + WS_LGK;
    float* bes = ws + WS_BES;
    float* mu  = ws + WS_MU;

    vmf_zero<<<(WS_N + KC + 255) / 256, 256, 0, stream>>>(ws, out);
    vmf_segsum<<<dim3(8, 32), 256, 0, stream>>>(X, y, r, n);
    vmf_stats<<<1, 128, 0, stream>>>(r, n, kap, lgk, bes, mu);
    vmf_kl<<<dim3(8, 8), 32, 0, stream>>>(mu, kap, lgk, bes, out);
}